// KingLoss_1958505087540
// MI455X (gfx1250) — compile-verified
//
#include <hip/hip_runtime.h>

// ---------------------------------------------------------------------------
// KingLoss on MI455X (gfx1250): HBM-bound streaming CE reduction.
// 185 MB of input @ 23.3 TB/s => ~8 us floor. Strategy: double-buffered
// async global->LDS staging (CDNA5 ASYNCcnt path) + register-resident
// per-row logsumexp + deterministic two-stage reduction.
// ---------------------------------------------------------------------------

#define KING 3
#define NCLS 10
#define THREADS 256
#define ROWS_PER_CHUNK 512
#define F4_PER_CHUNK (ROWS_PER_CHUNK * NCLS / 4)   // 1280 float4 per chunk
#define F4_PER_THREAD (F4_PER_CHUNK / THREADS)     // 5 b128 copies / thread
#define CHUNKS_PER_BLOCK 8

#define AS_GLOBAL __attribute__((address_space(1)))
#define AS_LDS    __attribute__((address_space(3)))

typedef int v4i __attribute__((vector_size(16)));

// ---- CDNA5 async global->LDS copy (tracked by ASYNCcnt) -------------------
// Probe-confirmed: builtin exists; param0 is 'v4i __device__*' (addrspace(1)).
#if __has_builtin(__builtin_amdgcn_global_load_async_to_lds_b128)
__device__ __forceinline__ void async_cp16(const float4* g, float4* l) {
  v4i* gg = (v4i*)(float4*)g;  // strip const in generic AS first
  __builtin_amdgcn_global_load_async_to_lds_b128(
      (AS_GLOBAL v4i*)gg, (AS_LDS v4i*)l, /*offset=*/0, /*cpol=*/0);
}
#else
__device__ __forceinline__ void async_cp16(const float4* g, float4* l) {
  unsigned loff = (unsigned)(unsigned long long)(AS_LDS void*)l;
  unsigned long long ga = (unsigned long long)(AS_GLOBAL const void*)g;
  asm volatile("global_load_async_to_lds_b128 %0, %1, off"
               :: "v"(loff), "v"(ga) : "memory");
}
#endif

#if __has_builtin(__builtin_amdgcn_s_wait_asynccnt)
#define WAIT_ASYNC(n) __builtin_amdgcn_s_wait_asynccnt(n)
#else
#define WAIT_ASYNC(n) asm volatile("s_wait_asynccnt " #n ::: "memory")
#endif

// Stage one 512-row chunk (20480 B) into LDS: 5 coalesced, aligned b128
// async copies per thread. Chunk base = chunk*512*10 floats, 16B-aligned.
__device__ __forceinline__ void stage_chunk(float4* lbuf, const float* gbase,
                                            long long chunk, int t) {
  const float4* g =
      (const float4*)(gbase + chunk * (long long)(ROWS_PER_CHUNK * NCLS));
#pragma unroll
  for (int k = 0; k < F4_PER_THREAD; ++k)
    async_cp16(g + t + k * THREADS, lbuf + t + k * THREADS);
}

__global__ __launch_bounds__(THREADS)
void king_loss_main(const float* __restrict__ logits,
                    const int* __restrict__ target,
                    const int* __restrict__ epoch,
                    float* __restrict__ partials) {
  __shared__ float4 buf[2][F4_PER_CHUNK];   // 2 x 20 KB double buffer
  __shared__ float red[THREADS];

  const int t = threadIdx.x;
  const bool kingEpoch = (epoch[0] % 5) == 0;
  const long long chunk0 = (long long)blockIdx.x * CHUNKS_PER_BLOCK;

  // Prologue: kick off chunk 0.
  stage_chunk(&buf[0][0], logits, chunk0, t);

  float acc = 0.f;
  for (int c = 0; c < CHUNKS_PER_BLOCK; ++c) {
    if (c + 1 < CHUNKS_PER_BLOCK) {
      // Prefetch next chunk, then retire the older 5 copies (in-order cnt).
      stage_chunk(&buf[(c + 1) & 1][0], logits, chunk0 + c + 1, t);
      WAIT_ASYNC(5);
    } else {
      WAIT_ASYNC(0);
    }
    __syncthreads();  // all waves' async data for chunk c now visible

    const float* lb = (const float*)&buf[c & 1][0];
    const long long rowBase = (chunk0 + c) * (long long)ROWS_PER_CHUNK;

#pragma unroll
    for (int r = 0; r < ROWS_PER_CHUNK / THREADS; ++r) {
      const int lrow = t + r * THREADS;
      // Conflict-free LDS reads: dword stride 10 is injective mod 64 banks
      // over the 32 lanes of a gfx1250 wave.
      float x[NCLS];
#pragma unroll
      for (int j = 0; j < NCLS; ++j) x[j] = lb[lrow * NCLS + j];

      float m = x[0];
#pragma unroll
      for (int j = 1; j < NCLS; ++j) m = fmaxf(m, x[j]);

      float s = 0.f;
#pragma unroll
      for (int j = 0; j < NCLS; ++j) s += __expf(x[j] - m);

      const int ti = target[rowBase + lrow];
      float xt = x[0];  // select-chain: keeps x[] in VGPRs (no scratch)
#pragma unroll
      for (int j = 1; j < NCLS; ++j) xt = (ti == j) ? x[j] : xt;

      float ce = m + __logf(s) - xt;
      if (kingEpoch) {
        if (ti != KING) ce += __expf(x[KING] - m) / s;
      } else {
        if (ti != KING) ce = 0.f;
      }
      acc += ce;
    }
    __syncthreads();  // buf[c&1] free to be overwritten by iteration c+1
  }

  // Deterministic in-block tree reduction.
  red[t] = acc;
  __syncthreads();
#pragma unroll
  for (int sft = THREADS / 2; sft > 0; sft >>= 1) {
    if (t < sft) red[t] += red[t + sft];
    __syncthreads();
  }
  if (t == 0) partials[blockIdx.x] = red[0];
}

__global__ __launch_bounds__(THREADS)
void king_loss_reduce(const float* __restrict__ partials, int n, float invN,
                      float* __restrict__ out) {
  __shared__ float red[THREADS];
  float a = 0.f;
  for (int i = threadIdx.x; i < n; i += THREADS) a += partials[i];
  red[threadIdx.x] = a;
  __syncthreads();
  for (int sft = THREADS / 2; sft > 0; sft >>= 1) {
    if (threadIdx.x < sft) red[threadIdx.x] += red[threadIdx.x + sft];
    __syncthreads();
  }
  if (threadIdx.x == 0) out[0] = red[0] * invN;
}

extern "C" void kernel_launch(void* const* d_in, const int* in_sizes, int n_in,
                              void* d_out, int out_size, void* d_ws, size_t ws_size,
                              hipStream_t stream) {
  const float* logits = (const float*)d_in[0];   // [N, 10] f32
  const int*   target = (const int*)d_in[1];     // [N] int (JAX default x32)
  const int*   epoch  = (const int*)d_in[2];     // scalar
  float* out = (float*)d_out;
  float* partials = (float*)d_ws;                // blocks * 4 bytes

  const long long N = (long long)in_sizes[0] / NCLS;          // 4194304
  const int nChunks = (int)(N / ROWS_PER_CHUNK);              // 8192
  const int blocks  = nChunks / CHUNKS_PER_BLOCK;             // 1024

  king_loss_main<<<blocks, THREADS, 0, stream>>>(logits, target, epoch, partials);
  king_loss_reduce<<<1, THREADS, 0, stream>>>(partials, blocks,
                                              (float)(1.0 / (double)N), out);
}